// TransE_19370302505582
// MI455X (gfx1250) — compile-verified
//
#include <hip/hip_runtime.h>
#include <stdint.h>

// TransE L1 score_tails: out[b,n] = -sum_k |(head[b]+rel[relid[b]])[k] - tail[n][k]|
// B=512, N=2048, D=128.
//
// MI455X plan:
//  - 64x64 output tile per 256-thread (8-wave) block; 4x4 register microtile.
//  - Tail tile moved global->LDS by the Tensor Data Mover (one descriptor DMA
//    per block, TENSORcnt), with TDM padding producing the 132-float row
//    stride for bank-conflict-free ds_load_b128.
//  - Query tile (head + gathered rel) built with VALU, overlapping the DMA.
//  - L1 (P=1) has no bilinear structure -> no WMMA path exists; compute core
//    is v_sub_f32 + v_add_f32 |src| = 2 VALU/element, VALU:LDS = 16:1.

#define D      128
#define BT     64          // batch tile
#define NT     64          // neg tile
#define PADW   4           // pad DWORDs per row
#define LSTR   (D + PADW)  // padded LDS row stride (floats) = 132
#define NTHR   256

typedef int          v4i  __attribute__((ext_vector_type(4)));
typedef unsigned int u32x4 __attribute__((ext_vector_type(4)));
typedef int          i32x4 __attribute__((ext_vector_type(4)));
typedef int          i32x8 __attribute__((ext_vector_type(8)));
typedef __attribute__((address_space(1))) v4i*   gbl_v4i_p;
typedef __attribute__((address_space(3))) v4i*   lds_v4i_p;
typedef __attribute__((address_space(3))) float* lds_f_p;

#if defined(__gfx1250__) && __has_builtin(__builtin_amdgcn_tensor_load_to_lds) && __has_builtin(__builtin_amdgcn_s_wait_tensorcnt)
  #define USE_TDM   1
#elif defined(__gfx1250__) && __has_builtin(__builtin_amdgcn_global_load_async_to_lds_b128)
  #define USE_ASYNC 1
#endif

__global__ __launch_bounds__(NTHR)
void transe_l1_tile(const float* __restrict__ head,
                    const int*   __restrict__ relid,
                    const float* __restrict__ tail,
                    const float* __restrict__ relemb,
                    float*       __restrict__ out,
                    int nneg)
{
    __shared__ float qs[BT * LSTR];   // query tile: head+rel
    __shared__ float ts[NT * LSTR];   // tail tile (row stride LSTR via TDM pad)

    const int tid = threadIdx.x;
    const int b0  = blockIdx.y * BT;
    const int n0  = blockIdx.x * NT;

    // ---------------- 1) move tail tile into LDS ----------------
#if USE_TDM
    // One TDM descriptor per block: 64x128 f32 tile, row stride 128 elems in
    // memory; pad 4 DWORDs after every 128 DWORDs in LDS -> 132-float rows.
    if (__builtin_amdgcn_readfirstlane(tid >> 5) == 0) {   // wave 0 only (TDM ignores EXEC)
        const uint64_t ga = (uint64_t)(uintptr_t)(tail + (size_t)n0 * D);
        const uint32_t la = (uint32_t)(uintptr_t)(lds_f_p)ts;   // LDS byte offset

        u32x4 g0;
        g0[0] = 1u;                                   // count=1, user D#, no gather
        g0[1] = la;                                   // lds_addr (bytes)
        g0[2] = (uint32_t)(ga & 0xffffffffu);         // global_addr[31:0]
        g0[3] = ((uint32_t)(ga >> 32) & 0x01ffffffu)  // global_addr[56:32]
                | (2u << 30);                         // type = 2 ("image")

        i32x8 g1;
        g1[0] = (int)((2u << 16)      // data_size = 4 bytes
                    | (1u << 20)      // pad_enable
                    | (6u << 22)      // pad_interval: 128 DWORDs (=512B = one row)
                    | (3u << 25));    // pad_amount: 4 DWORDs (=16B)
        g1[1] = (int)((uint32_t)D  << 16);   // tensor_dim0[15:0] = 128 (bar_addr=0)
        g1[2] = (int)((uint32_t)NT << 16);   // tensor_dim0 hi=0 | tensor_dim1 lo=64
        g1[3] = (int)((uint32_t)D  << 16);   // tensor_dim1 hi=0 | tile_dim0 = 128
        g1[4] = NT;                          // tile_dim1 = 64, tile_dim2 = 0 (2D)
        g1[5] = D;                           // tensor_dim0_stride[31:0] = 128
        g1[6] = 0;                           // stride0 hi | tensor_dim1_stride lo
        g1[7] = 0;                           // tensor_dim1_stride hi

        const i32x4 gz4 = {0, 0, 0, 0};              // groups 2/3 unused (tile_dim2=0)
        const i32x8 gz8 = {0, 0, 0, 0, 0, 0, 0, 0};  // trailing words (6-arg form)
        __builtin_amdgcn_tensor_load_to_lds(g0, g1, gz4, gz4, gz8, /*cpol=*/0);
    }
#elif USE_ASYNC
    #pragma unroll
    for (int s = 0; s < (NT * D / 4) / NTHR; ++s) {
        const int c    = tid + NTHR * s;
        const int row  = c >> 5;          // 32 float4 per row
        const int col4 = (c & 31) << 2;
        const float* g = tail + (size_t)(n0 + row) * D + col4;
        float*       l = ts + row * LSTR + col4;
        __builtin_amdgcn_global_load_async_to_lds_b128((gbl_v4i_p)g, (lds_v4i_p)l, 0, 0);
    }
#else
    #pragma unroll
    for (int s = 0; s < (NT * D / 4) / NTHR; ++s) {
        const int c    = tid + NTHR * s;
        const int row  = c >> 5;
        const int col4 = (c & 31) << 2;
        const float4 v = *(const float4*)(tail + (size_t)(n0 + row) * D + col4);
        *(float4*)(ts + row * LSTR + col4) = v;
    }
#endif

    // ------- 2) build query tile (head + gather(rel)) — overlaps the DMA -------
    #pragma unroll
    for (int s = 0; s < (BT * D / 4) / NTHR; ++s) {
        const int c    = tid + NTHR * s;
        const int row  = c >> 5;
        const int col4 = (c & 31) << 2;
        const int b    = b0 + row;
        const int r    = relid[b];
        const float4 h  = *(const float4*)(head   + (size_t)b * D + col4);
        const float4 re = *(const float4*)(relemb + (size_t)r * D + col4);
        float4 q;
        q.x = h.x + re.x; q.y = h.y + re.y; q.z = h.z + re.z; q.w = h.w + re.w;
        *(float4*)(qs + row * LSTR + col4) = q;
    }

#if USE_TDM
    __builtin_amdgcn_s_wait_tensorcnt(0);    // issuing wave drains TDM; others trivial
#elif USE_ASYNC
    __builtin_amdgcn_s_wait_asynccnt(0);
#endif
    __syncthreads();

    // ---------------- 3) 4x4 register microtile over K ----------------
    const int tx = tid & 15;   // neg lane   (rows tx, tx+16, tx+32, tx+48)
    const int ty = tid >> 4;   // batch lane (rows ty, ty+16, ty+32, ty+48)

    float acc[4][4];
    #pragma unroll
    for (int i = 0; i < 4; ++i)
        #pragma unroll
        for (int j = 0; j < 4; ++j) acc[i][j] = 0.0f;

    #pragma unroll 2
    for (int k = 0; k < D; k += 4) {
        float4 qv[4], tv[4];
        #pragma unroll
        for (int i = 0; i < 4; ++i)
            qv[i] = *(const float4*)(qs + (ty + 16 * i) * LSTR + k);
        #pragma unroll
        for (int j = 0; j < 4; ++j)
            tv[j] = *(const float4*)(ts + (tx + 16 * j) * LSTR + k);

        #pragma unroll
        for (int i = 0; i < 4; ++i)
            #pragma unroll
            for (int j = 0; j < 4; ++j) {
                acc[i][j] += fabsf(qv[i].x - tv[j].x)
                           + fabsf(qv[i].y - tv[j].y)
                           + fabsf(qv[i].z - tv[j].z)
                           + fabsf(qv[i].w - tv[j].w);
            }
    }

    // ---------------- 4) write scores ----------------
    #pragma unroll
    for (int i = 0; i < 4; ++i) {
        const size_t b = b0 + ty + 16 * i;
        #pragma unroll
        for (int j = 0; j < 4; ++j) {
            const int n = n0 + tx + 16 * j;
            out[b * (size_t)nneg + n] = -acc[i][j];
        }
    }
}

extern "C" void kernel_launch(void* const* d_in, const int* in_sizes, int n_in,
                              void* d_out, int out_size, void* d_ws, size_t ws_size,
                              hipStream_t stream)
{
    const float* head   = (const float*)d_in[0];   // (B, 128) f32
    const int*   relid  = (const int*)  d_in[1];   // (B,) int
    const float* tail   = (const float*)d_in[2];   // (1, N, 128) f32
    const float* relemb = (const float*)d_in[3];   // (n_rel, 128) f32
    float*       out    = (float*)d_out;           // (B, N) f32

    const int batch = in_sizes[0] / D;   // 512
    const int nneg  = in_sizes[2] / D;   // 2048

    dim3 grid(nneg / NT, batch / BT);    // 32 x 8 = 256 blocks
    transe_l1_tile<<<grid, NTHR, 0, stream>>>(head, relid, tail, relemb, out, nneg);
}